// InplaceUpsampler_22170621182468
// MI455X (gfx1250) — compile-verified
//
#include <hip/hip_runtime.h>

// Problem constants (from reference): x [32,256,64,64] f32, out [32,64,128,128] f32
#define B_N   32
#define C_IN  256
#define C4    64
#define Hc    64
#define Wc    64
#define H2    128
#define W2    128

// Coarse tile per workgroup
#define TW 32
#define TH 8
#define PW (TW + 2)          // 34 (patch width with halo)
#define PH (TH + 2)          // 10 (patch height with halo)
#define PLANE (PW * PH)      // 340 floats per plane
#define NLOAD (4 * PLANE)    // 1360 floats total

// ---- CDNA5 async global->LDS path (guarded; falls back to plain loads) ----
#ifdef __has_builtin
#  if defined(__AMDGCN__) && __has_builtin(__builtin_amdgcn_global_load_async_to_lds_b32)
#    define USE_ASYNC_LDS 1
#  endif
#  if defined(__AMDGCN__) && __has_builtin(__builtin_amdgcn_s_wait_asynccnt)
#    define HAVE_WAIT_ASYNC_BUILTIN 1
#  endif
#endif

#if defined(USE_ASYNC_LDS)
typedef __attribute__((address_space(1))) int* gint_ptr;
typedef __attribute__((address_space(3))) int* lint_ptr;
#endif

__global__ __launch_bounds__(256)
void InplaceUpsampler_22170621182468_kernel(const float* __restrict__ x,
                                            float* __restrict__ out)
{
    __shared__ float patch[NLOAD];

    const int tid = threadIdx.x;
    const int tx0 = (blockIdx.x & 1) * TW;        // coarse tile origin (x)
    const int ty0 = (blockIdx.x >> 1) * TH;       // coarse tile origin (y)
    const int cp  = blockIdx.y;                   // output channel c' in [0,64)
    const int b   = blockIdx.z;                   // batch

    // plane p (= 2i+j) lives at input channel p*C4 + cp
    const float* xb = x + ((size_t)(b * C_IN + cp) * Hc) * Wc;

    // ---------------- Load 4 halo patches into LDS ----------------
    for (int idx = tid; idx < NLOAD; idx += 256) {
        int p   = idx / PLANE;
        int rem = idx - p * PLANE;
        int r   = rem / PW;
        int c   = rem - r * PW;
        int gh  = ty0 + r - 1;
        int gw  = tx0 + c - 1;
        bool valid = ((unsigned)gh < (unsigned)Hc) && ((unsigned)gw < (unsigned)Wc);
#if defined(USE_ASYNC_LDS)
        if (valid) {
            const float* gp = xb + (size_t)p * (C4 * Hc * Wc) + gh * Wc + gw;
            __builtin_amdgcn_global_load_async_to_lds_b32(
                (gint_ptr)gp, (lint_ptr)&patch[idx], 0, 0);
        } else {
            patch[idx] = 0.0f;                    // zero halo via ds_store (disjoint addrs)
        }
#else
        const float* gp = xb + (size_t)p * (C4 * Hc * Wc) + gh * Wc + gw;
        patch[idx] = valid ? gp[0] : 0.0f;
#endif
    }

#if defined(USE_ASYNC_LDS)
#  if defined(HAVE_WAIT_ASYNC_BUILTIN)
    __builtin_amdgcn_s_wait_asynccnt(0);
#  else
    asm volatile("s_wait_asynccnt 0" ::: "memory");
#  endif
#endif
    __syncthreads();

    // ---------------- Compute one 2x2 output quad per thread ----------------
    const int tx = tid & (TW - 1);                // 0..31
    const int ty = tid >> 5;                      // 0..7
    const int r0 = ty + 1;
    const int c0 = tx + 1;

    // fine index fr (rows 2h-1..2h+2) -> (coarse delta, parity i')
    const int dmap[4] = { -1, 0, 0, 1 };
    const int imap[4] = {  1, 0, 1, 0 };

    float F[4][4];
#pragma unroll
    for (int fr = 0; fr < 4; ++fr) {
#pragma unroll
        for (int fc = 0; fc < 4; ++fc) {
            int p = 2 * imap[fr] + imap[fc];      // plane = 2i'+j'
            F[fr][fc] = patch[p * PLANE + (r0 + dmap[fr]) * PW + (c0 + dmap[fc])];
        }
    }

    // separable [1,2,1] vertical then horizontal, /16
    float G0[4], G1[4];
#pragma unroll
    for (int fc = 0; fc < 4; ++fc) {
        G0[fc] = F[0][fc] + 2.0f * F[1][fc] + F[2][fc];
        G1[fc] = F[1][fc] + 2.0f * F[2][fc] + F[3][fc];
    }
    const float s = 0.0625f;
    float o00 = (G0[0] + 2.0f * G0[1] + G0[2]) * s;
    float o01 = (G0[1] + 2.0f * G0[2] + G0[3]) * s;
    float o10 = (G1[0] + 2.0f * G1[1] + G1[2]) * s;
    float o11 = (G1[1] + 2.0f * G1[2] + G1[3]) * s;

    const int gh = ty0 + ty;
    const int gw = tx0 + tx;
    size_t obase = (((size_t)(b * C4 + cp) * H2) + 2 * gh) * W2 + 2 * gw;
    *(float2*)(out + obase)      = make_float2(o00, o01);
    *(float2*)(out + obase + W2) = make_float2(o10, o11);
}

extern "C" void kernel_launch(void* const* d_in, const int* in_sizes, int n_in,
                              void* d_out, int out_size, void* d_ws, size_t ws_size,
                              hipStream_t stream) {
    (void)in_sizes; (void)n_in; (void)d_ws; (void)ws_size; (void)out_size;
    const float* x  = (const float*)d_in[0];   // [32,256,64,64] f32
    // d_in[1] is the 3x3 kernel; it is the fixed binomial [1,2,1]^T[1,2,1]/16, hardcoded above.
    float* out = (float*)d_out;                // [32,64,128,128] f32

    dim3 grid(16, C4, B_N);                    // 16 tiles (2x8) per (b, c')
    dim3 block(256);
    InplaceUpsampler_22170621182468_kernel<<<grid, block, 0, stream>>>(x, out);
}